// GammaCapsule_51754355917245
// MI455X (gfx1250) — compile-verified
//
#include <hip/hip_runtime.h>

#define BATCH   64
#define INCAPS  1152
#define EDIM    8
#define CAPS    64
#define DDIM    16
#define NBLK    (BATCH * CAPS)          // 4096 routing blocks
// log(0.9*(caps-1)) - log(1-0.9) = log(567)
#define LOG_PRIOR 6.3403594f

typedef __attribute__((ext_vector_type(2))) float v2f;
typedef __attribute__((ext_vector_type(8))) float v8f;

// ---------------------------------------------------------------------------
// u_norm[b,i] = ||u[b,i,:]|| over E=8
__global__ __launch_bounds__(256) void unorm_kernel(const float* __restrict__ u,
                                                    float* __restrict__ unorm) {
    const int idx = blockIdx.x * 256 + threadIdx.x;
    if (idx < BATCH * INCAPS) {
        const float* p = u + (size_t)idx * EDIM;
        float n2 = 0.0f;
        #pragma unroll
        for (int e = 0; e < EDIM; ++e) n2 += p[e] * p[e];
        unorm[idx] = sqrtf(n2);
    }
}

// ---------------------------------------------------------------------------
// u_hat[b,c,i,d] = sum_e u[b,i,e] * W[c,i,d,e]
// Per (c,i): D[16b x 16d] = A[16b x 8e] * B[8e x 16d] as two chained
// V_WMMA_F32_16X16X4_F32 (K=0..3 then K=4..7). One wave per (b-tile, c, i).
__global__ __launch_bounds__(256) void uhat_wmma_kernel(const float* __restrict__ u,
                                                        const float* __restrict__ W,
                                                        float* __restrict__ uhat) {
    const int lane = threadIdx.x & 31;
    const int wid  = blockIdx.x * 8 + (threadIdx.x >> 5);   // global wave id
    const int bt   = wid & 3;                                // batch tile 0..3
    const int t2   = wid >> 2;
    const int i    = t2 % INCAPS;
    const int c    = t2 / INCAPS;
    const int half = lane >> 4;                              // 0 or 1
    const int mn   = lane & 15;                              // M for A, N for B/D

    // A: u[b, i, e] ; this lane supplies e = 2*half + {0,1} (and +4 for 2nd wmma)
    const int b = bt * 16 + mn;
    const float* ub = u + ((size_t)b * INCAPS + i) * EDIM + 2 * half;
    v2f a0 = { ub[0], ub[1] };
    v2f a1 = { ub[4], ub[5] };

    // B: B[e][d] = W[c, i, d, e] ; lane supplies column d=mn, rows e=2*half+{0,1}
    const float* wb = W + (((size_t)c * INCAPS + i) * DDIM + mn) * EDIM + 2 * half;
    v2f bv0 = { wb[0], wb[1] };
    v2f bv1 = { wb[4], wb[5] };

    v8f acc = {0.f, 0.f, 0.f, 0.f, 0.f, 0.f, 0.f, 0.f};
    acc = __builtin_amdgcn_wmma_f32_16x16x4_f32(false, a0, false, bv0, (short)0, acc, false, false);
    acc = __builtin_amdgcn_wmma_f32_16x16x4_f32(false, a1, false, bv1, (short)0, acc, false, false);

    #pragma unroll
    for (int j = 0; j < 8; ++j) {
        const int row = bt * 16 + j + 8 * half;              // batch index
        uhat[(((size_t)row * CAPS + c) * INCAPS + i) * DDIM + mn] = acc[j];
    }
}

// ---------------------------------------------------------------------------
// One routing iteration for one (b,c) pair per block.
// LDS tile stored SoA (uh[d][i]) so wave lanes (consecutive i) hit distinct
// banks -> conflict-free on all hot loops. Block partial sums of d go to
// bsums[] (deterministic two-stage reduction instead of float atomics).
template <int PHASE>
__global__ __launch_bounds__(256) void route_kernel(const float* __restrict__ uhat,
                                                    const float* __restrict__ unorm,
                                                    const float* __restrict__ bias,
                                                    const float* __restrict__ d_prev,
                                                    const float* __restrict__ denom,
                                                    const float* __restrict__ tval,
                                                    float* __restrict__ d_cur,
                                                    float* __restrict__ bsums,
                                                    float* __restrict__ v_out,
                                                    float* __restrict__ c_out) {
    extern __shared__ float smem[];
    float* uh  = smem;                          // [DDIM][INCAPS]
    float* red = smem + DDIM * INCAPS;          // [DDIM][256]
    float* vsh = red + DDIM * 256;              // [DDIM]

    const int tid = threadIdx.x;
    const int c   = blockIdx.x;
    const int b   = blockIdx.y;

    // Load u_hat tile, fold in the norm clip: scale = min(||row||, ||u||)/||row||
    for (int i = tid; i < INCAPS; i += 256) {
        const float4* row = (const float4*)(uhat + (((size_t)b * CAPS + c) * INCAPS + i) * DDIM);
        float r[DDIM];
        float n2 = 0.0f;
        #pragma unroll
        for (int k = 0; k < 4; ++k) {
            float4 q = row[k];
            r[4*k+0] = q.x; r[4*k+1] = q.y; r[4*k+2] = q.z; r[4*k+3] = q.w;
        }
        #pragma unroll
        for (int d = 0; d < DDIM; ++d) n2 += r[d] * r[d];
        const float uhn = sqrtf(n2);
        const float un  = unorm[(size_t)b * INCAPS + i];
        const float sc  = fminf(uhn, un) / fmaxf(uhn, 1e-12f);
        #pragma unroll
        for (int d = 0; d < DDIM; ++d) uh[d * INCAPS + i] = r[d] * sc;
    }
    __syncthreads();

    float t = 0.0f;
    if (PHASE > 0) t = tval[0];

    // s_j = sum_i c_ij * u_hat_i  (partial per thread, then LDS tree reduce)
    float ps[DDIM];
    #pragma unroll
    for (int d = 0; d < DDIM; ++d) ps[d] = 0.0f;

    for (int i = tid; i < INCAPS; i += 256) {
        float cij;
        if (PHASE == 0) {
            cij = 1.0f / (float)CAPS;            // softmax of zeros
        } else {
            const float bij = t * d_prev[(((size_t)b * CAPS + c) * INCAPS + i)];
            cij = __expf(bij) / denom[(size_t)b * INCAPS + i];
        }
        if (PHASE == 2) c_out[((size_t)b * CAPS + c) * INCAPS + i] = cij;
        #pragma unroll
        for (int d = 0; d < DDIM; ++d) ps[d] += cij * uh[d * INCAPS + i];
    }
    #pragma unroll
    for (int d = 0; d < DDIM; ++d) red[d * 256 + tid] = ps[d];
    __syncthreads();
    for (int off = 128; off > 0; off >>= 1) {
        if (tid < off) {
            #pragma unroll
            for (int d = 0; d < DDIM; ++d)
                red[d * 256 + tid] += red[d * 256 + tid + off];
        }
        __syncthreads();
    }

    // squash
    if (tid == 0) {
        float s[DDIM];
        float n2 = 0.0f;
        #pragma unroll
        for (int d = 0; d < DDIM; ++d) {
            s[d] = red[d * 256] + bias[c * DDIM + d];
            n2 += s[d] * s[d];
        }
        const float f = (n2 / (1.0f + n2)) / sqrtf(n2 + 1e-7f);
        #pragma unroll
        for (int d = 0; d < DDIM; ++d) vsh[d] = s[d] * f;
    }
    __syncthreads();

    if (PHASE == 2) {
        if (tid < DDIM) v_out[((size_t)b * CAPS + c) * DDIM + tid] = vsh[tid];
        return;
    }

    // d_i = ||v - u_hat_i|| ; store + block partial sum (deterministic)
    float lsum = 0.0f;
    for (int i = tid; i < INCAPS; i += 256) {
        float n2 = 0.0f;
        #pragma unroll
        for (int d = 0; d < DDIM; ++d) {
            const float df = vsh[d] - uh[d * INCAPS + i];
            n2 += df * df;
        }
        const float dv = sqrtf(n2);
        d_cur[(((size_t)b * CAPS + c) * INCAPS + i)] = dv;
        lsum += dv;
    }
    red[tid] = lsum;
    __syncthreads();
    for (int off = 128; off > 0; off >>= 1) {
        if (tid < off) red[tid] += red[tid + off];
        __syncthreads();
    }
    if (tid == 0) bsums[b * CAPS + c] = red[0];
}

// ---------------------------------------------------------------------------
// Deterministically sum the 4096 block partials and emit
// t = log_prior / (-0.5*mean(d) + 1e-12)
__global__ __launch_bounds__(256) void reduce_t_kernel(const float* __restrict__ bsums,
                                                       float* __restrict__ tval) {
    __shared__ float red[256];
    const int tid = threadIdx.x;
    float s = 0.0f;
    for (int k = tid; k < NBLK; k += 256) s += bsums[k];   // fixed order
    red[tid] = s;
    __syncthreads();
    for (int off = 128; off > 0; off >>= 1) {
        if (tid < off) red[tid] += red[tid + off];
        __syncthreads();
    }
    if (tid == 0) {
        const float d_o = red[0] * (1.0f / (float)((size_t)BATCH * CAPS * INCAPS));
        tval[0] = LOG_PRIOR / (-0.5f * d_o + 1e-12f);
    }
}

// ---------------------------------------------------------------------------
// denom[b,i] = sum_c exp(t * d[b,c,i])  (softmax over output capsules)
__global__ __launch_bounds__(256) void denom_kernel(const float* __restrict__ dbuf,
                                                    const float* __restrict__ tval,
                                                    float* __restrict__ denom) {
    const int idx = blockIdx.x * 256 + threadIdx.x;
    if (idx < BATCH * INCAPS) {
        const float t = tval[0];
        const int b = idx / INCAPS;
        const int i = idx % INCAPS;
        float s = 0.0f;
        #pragma unroll 4
        for (int c = 0; c < CAPS; ++c)
            s += __expf(t * dbuf[(((size_t)b * CAPS + c) * INCAPS + i)]);
        denom[idx] = s;
    }
}

// ---------------------------------------------------------------------------
extern "C" void kernel_launch(void* const* d_in, const int* in_sizes, int n_in,
                              void* d_out, int out_size, void* d_ws, size_t ws_size,
                              hipStream_t stream) {
    const float* u    = (const float*)d_in[0];   // (64,1152,8)
    const float* W    = (const float*)d_in[1];   // (1,64,1152,16,8)
    const float* bias = (const float*)d_in[2];   // (1,64,16)
    float* out = (float*)d_out;
    float* ws  = (float*)d_ws;

    const size_t N_UHAT = (size_t)BATCH * CAPS * INCAPS * DDIM;  // 75,497,472
    const size_t N_BI   = (size_t)BATCH * INCAPS;                //     73,728
    const size_t N_D    = (size_t)BATCH * CAPS * INCAPS;         //  4,718,592

    float* uhat  = ws;
    float* unorm = uhat  + N_UHAT;
    float* dbuf0 = unorm + N_BI;
    float* dbuf1 = dbuf0 + N_D;
    float* denom = dbuf1 + N_D;
    float* bsums = denom + N_BI;   // [NBLK]
    float* tvals = bsums + NBLK;   // [2]

    float* v_out = out;                                   // (64,64,16)
    float* c_out = out + (size_t)BATCH * CAPS * DDIM;     // (64,64,1152)

    unorm_kernel<<<(BATCH * INCAPS + 255) / 256, 256, 0, stream>>>(u, unorm);

    // 4 b-tiles * 64 c * 1152 i waves, 8 waves per 256-thread block
    uhat_wmma_kernel<<<(4 * CAPS * INCAPS) / 8, 256, 0, stream>>>(u, W, uhat);

    const size_t smem = (size_t)(DDIM * INCAPS + DDIM * 256 + DDIM) * sizeof(float);
    dim3 rg(CAPS, BATCH);
    route_kernel<0><<<rg, 256, smem, stream>>>(uhat, unorm, bias, nullptr, nullptr, nullptr,
                                               dbuf0, bsums, nullptr, nullptr);
    reduce_t_kernel<<<1, 256, 0, stream>>>(bsums, tvals + 0);
    denom_kernel<<<(BATCH * INCAPS + 255) / 256, 256, 0, stream>>>(dbuf0, tvals + 0, denom);
    route_kernel<1><<<rg, 256, smem, stream>>>(uhat, unorm, bias, dbuf0, denom, tvals + 0,
                                               dbuf1, bsums, nullptr, nullptr);
    reduce_t_kernel<<<1, 256, 0, stream>>>(bsums, tvals + 1);
    denom_kernel<<<(BATCH * INCAPS + 255) / 256, 256, 0, stream>>>(dbuf1, tvals + 1, denom);
    route_kernel<2><<<rg, 256, smem, stream>>>(uhat, unorm, bias, dbuf1, denom, tvals + 1,
                                               nullptr, nullptr, v_out, c_out);
}